// SignatureTransform_37383395344448
// MI455X (gfx1250) — compile-verified
//
#include <hip/hip_runtime.h>

// ---------------------------------------------------------------------------
// Depth-4 path signature of an [8192, 8] float32 path.
// Parallelized via associativity of Chen's relation:
//   phase 1: per-chunk signatures (256 chunks x 32 increments) in LDS
//   phase 2: exclusive prefix scan of levels 1..3 across chunks (584 floats)
//   phase 3: level-4/3/2 cross terms as K=256 GEMMs via v_wmma_f32_16x16x4_f32
//            (A/B panels staged in LDS, branch-free inner loop)
//   phase 4: deterministic final reduction into d_out
// ---------------------------------------------------------------------------

#define NPOINTS 8192
#define NINC    (NPOINTS - 1)   // 8191
#define CHUNK   32
#define NCHUNK  256             // ceil(8191/32)
#define SIGLEN  4680            // 8 + 64 + 512 + 4096
#define L1OFF   0
#define L2OFF   8
#define L3OFF   72
#define L4OFF   584
#define PLEN    584             // levels 1..3 only
#define KSTRIDE 260             // LDS row stride: %64==4 -> conflict-free b64

typedef float v2f __attribute__((ext_vector_type(2)));
typedef float v8f __attribute__((ext_vector_type(8)));

// ---------------------------------------------------------------------------
// Phase 1: each block computes the depth-4 signature of its 32-increment chunk.
// Signature state (4680 f32) lives in LDS; 512 threads: thread t owns
// level-3 element t = (i,j,k) and level-4 elements t*8+l, threads <64 own
// level 2, threads <8 own level 1. Updates use the factorization
//   c4[ijk,l] += (a3 + a2*dk/2 + a1*dj*dk/6 + di*dj*dk/24) * dl
// Starting from the zero (identity) signature makes step 0 == exp(dx0).
// ---------------------------------------------------------------------------
__global__ __launch_bounds__(512)
void sig_chunk_kernel(const float* __restrict__ path,
                      float* __restrict__ chunkSig)
{
    __shared__ float sig[SIGLEN];
    __shared__ float pts[(CHUNK + 1) * 8];

    const int t = threadIdx.x;
    const int g = blockIdx.x;
    const int start = g * CHUNK;
    int cnt = NINC - start;
    if (cnt > CHUNK) cnt = CHUNK;

    for (int e = t; e < (cnt + 1) * 8; e += 512)
        pts[e] = path[start * 8 + e];
    for (int e = t; e < SIGLEN; e += 512)
        sig[e] = 0.0f;
    __syncthreads();

    const int i = t >> 6;
    const int j = (t >> 3) & 7;
    const int k = t & 7;

    for (int s = 0; s < cnt; ++s) {
        float dx[8];
#pragma unroll
        for (int c = 0; c < 8; ++c)
            dx[c] = pts[(s + 1) * 8 + c] - pts[s * 8 + c];

        // ---- read old state (cross-thread reads) ----
        const float a3  = sig[L3OFF + t];
        const float a2  = sig[L2OFF + (t >> 3)];
        const float a1  = sig[L1OFF + i];
        const float a1b = (t < 64) ? sig[L1OFF + (t >> 3)] : 0.0f;

        const float di = dx[i], dj = dx[j], dk = dx[k];
        const float s4 = a3 + 0.5f * a2 * dk
                            + (1.0f / 6.0f)  * a1 * dj * dk
                            + (1.0f / 24.0f) * di * dj * dk;
        const float s3 = (a2 + 0.5f * a1 * dj + (1.0f / 6.0f) * di * dj) * dk;
        const float s2 = (t < 64) ? (a1b + 0.5f * dx[t >> 3]) * dx[t & 7] : 0.0f;

        __syncthreads();   // all reads done before any write

        // ---- write new state (each element owned by exactly one thread) ----
#pragma unroll
        for (int l = 0; l < 8; ++l)
            sig[L4OFF + t * 8 + l] += s4 * dx[l];
        sig[L3OFF + t] += s3;
        if (t < 64) sig[L2OFF + t] += s2;
        if (t < 8)  sig[L1OFF + t] += dx[t];

        __syncthreads();
    }

    for (int e = t; e < SIGLEN; e += 512)
        chunkSig[g * SIGLEN + e] = sig[e];
}

// ---------------------------------------------------------------------------
// Phase 2: exclusive prefix scan of chunk signatures, levels 1..3 only
// (584 floats of running state).  P^{d+1} = Chen(P^d, A^d):
//   P3 += P2 (x) a1 + P1 (x) a2 + a3 ;  P2 += P1 (x) a1 + a2 ;  P1 += a1
// Writes the exclusive prefix P^d (before the update) for every d.
// ---------------------------------------------------------------------------
__global__ __launch_bounds__(256)
void prefix_scan_kernel(const float* __restrict__ chunkSig,
                        float* __restrict__ Pall)
{
    __shared__ float P[PLEN];
    const int t = threadIdx.x;
    for (int e = t; e < PLEN; e += 256) P[e] = 0.0f;
    __syncthreads();

    for (int d = 0; d < NCHUNK; ++d) {
        // store exclusive prefix
        for (int e = t; e < PLEN; e += 256)
            Pall[d * PLEN + e] = P[e];

        const float* A = chunkSig + d * SIGLEN;

        float n3[2];
#pragma unroll
        for (int u = 0; u < 2; ++u) {
            const int m  = t + u * 256;       // level-3 flat index (i,j,k)
            const int i  = m >> 6;
            const int ij = m >> 3;
            const int jk = m & 63;
            const int k  = m & 7;
            n3[u] = P[L3OFF + m]
                  + P[L2OFF + ij] * A[L1OFF + k]
                  + P[L1OFF + i]  * A[L2OFF + jk]
                  + A[L3OFF + m];
        }
        float n2 = 0.0f, n1 = 0.0f;
        if (t < 64) {
            const int i = t >> 3, j = t & 7;
            n2 = P[L2OFF + t] + P[L1OFF + i] * A[L1OFF + j] + A[L2OFF + t];
        }
        if (t < 8) n1 = P[L1OFF + t] + A[L1OFF + t];

        __syncthreads();
#pragma unroll
        for (int u = 0; u < 2; ++u) P[L3OFF + t + u * 256] = n3[u];
        if (t < 64) P[L2OFF + t] = n2;
        if (t < 8)  P[L1OFF + t] = n1;
        __syncthreads();
    }
}

// ---------------------------------------------------------------------------
// Phase 3: cross terms as GEMMs over K = NCHUNK chunks with f32 WMMA.
//   C[m,n] = sum_d A(m,d) * B(d,n),  A(m,d) = Pall[d][Aoff+m], B = chunkSig.
// One wave per 16x16 output tile.  A/B panels are staged K-major into LDS
// (out-of-range rows zero-filled there), so the 64-step K loop is branch-free:
// 2x ds_load_b64 + v_wmma_f32_16x16x4_f32 per step, EXEC all-ones throughout.
// A 16x4 f32 layout (ISA 7.12.2): lanes 0-15 -> M=lane, VGPR0=K0, VGPR1=K1;
// lanes 16-31 -> M=lane-16, K2/K3.  B 4x16 symmetric.  C/D: VGPR r holds
// (M=r, N=lane) for lanes<16 and (M=r+8, N=lane-16) for lanes>=16.
// ---------------------------------------------------------------------------
__global__ __launch_bounds__(32)
void combine_wmma_kernel(const float* __restrict__ Pall,
                         const float* __restrict__ chunkSig,
                         float* __restrict__ T1, float* __restrict__ T2,
                         float* __restrict__ T3, float* __restrict__ T4,
                         float* __restrict__ T5, float* __restrict__ T6)
{
    __shared__ float Ap[16 * KSTRIDE];   // Ap[m][k], K-major, padded rows
    __shared__ float Bp[16 * KSTRIDE];   // Bp[n][k]

    const int b    = blockIdx.x;
    const int lane = threadIdx.x;

    int Aoff, Mreal, Boff, Nreal, m0, n0;
    float* Cbuf;
    if (b < 32)      { Aoff=L3OFF; Mreal=512; Boff=L1OFF; Nreal=8;   m0=16*b;           n0=0;             Cbuf=T1; } // P3 (x) A1 -> L4
    else if (b < 48) { Aoff=L2OFF; Mreal=64;  Boff=L2OFF; Nreal=64;  m0=16*((b-32)>>2); n0=16*((b-32)&3); Cbuf=T2; } // P2 (x) A2 -> L4
    else if (b < 80) { Aoff=L1OFF; Mreal=8;   Boff=L3OFF; Nreal=512; m0=0;              n0=16*(b-48);     Cbuf=T3; } // P1 (x) A3 -> L4
    else if (b < 84) { Aoff=L2OFF; Mreal=64;  Boff=L1OFF; Nreal=8;   m0=16*(b-80);      n0=0;             Cbuf=T4; } // P2 (x) A1 -> L3
    else if (b < 88) { Aoff=L1OFF; Mreal=8;   Boff=L2OFF; Nreal=64;  m0=0;              n0=16*(b-84);     Cbuf=T5; } // P1 (x) A2 -> L3
    else             { Aoff=L1OFF; Mreal=8;   Boff=L1OFF; Nreal=8;   m0=0;              n0=0;             Cbuf=T6; } // P1 (x) A1 -> L2

    // ---- stage panels into LDS (row index m uniform per iteration) ----
    for (int idx = lane; idx < 16 * NCHUNK; idx += 32) {
        const int m = idx >> 8;             // 0..15 within tile
        const int k = idx & (NCHUNK - 1);   // 0..255
        const int gm = m0 + m;
        const int gn = n0 + m;
        Ap[m * KSTRIDE + k] = (gm < Mreal) ? Pall[k * PLEN + Aoff + gm] : 0.0f;
        Bp[m * KSTRIDE + k] = (gn < Nreal) ? chunkSig[k * SIGLEN + Boff + gn] : 0.0f;
    }
    __syncthreads();

    const int l15 = lane & 15;
    const int hi  = lane >> 4;              // lane half selects K pair
    const float* Arow = &Ap[l15 * KSTRIDE];
    const float* Brow = &Bp[l15 * KSTRIDE];

    v8f acc = {0.0f, 0.0f, 0.0f, 0.0f, 0.0f, 0.0f, 0.0f, 0.0f};

#pragma unroll 8
    for (int k0 = 0; k0 < NCHUNK; k0 += 4) {
        const int ka = k0 + 2 * hi;         // even -> 8B-aligned b64 loads
        const v2f a  = *(const v2f*)(Arow + ka);
        const v2f bv = *(const v2f*)(Brow + ka);
        acc = __builtin_amdgcn_wmma_f32_16x16x4_f32(
                  /*neg_a=*/false, a, /*neg_b=*/false, bv,
                  /*c_mod=*/(short)0, acc, /*reuse_a=*/false, /*reuse_b=*/false);
    }

    const int n     = n0 + l15;
    const int mbase = m0 + 8 * hi;
#pragma unroll
    for (int r = 0; r < 8; ++r) {
        const int m = mbase + r;
        if (m < Mreal && n < Nreal)
            Cbuf[m * Nreal + n] = acc[r];
    }
}

// ---------------------------------------------------------------------------
// Phase 4: out[e] = sum_d chunkSig[d][e]  +  GEMM partials for its level.
// Sequential per-thread sums -> bitwise deterministic (no atomics).
// ---------------------------------------------------------------------------
__global__ __launch_bounds__(256)
void finalize_kernel(const float* __restrict__ chunkSig,
                     const float* __restrict__ T1, const float* __restrict__ T2,
                     const float* __restrict__ T3, const float* __restrict__ T4,
                     const float* __restrict__ T5, const float* __restrict__ T6,
                     float* __restrict__ out)
{
    const int e = blockIdx.x * 256 + threadIdx.x;
    if (e >= SIGLEN) return;
    float s = 0.0f;
    for (int d = 0; d < NCHUNK; ++d)
        s += chunkSig[d * SIGLEN + e];
    if (e < L2OFF)        out[e] = s;
    else if (e < L3OFF)   out[e] = s + T6[e - L2OFF];
    else if (e < L4OFF) { const int x = e - L3OFF; out[e] = s + T4[x] + T5[x]; }
    else                { const int x = e - L4OFF; out[e] = s + T1[x] + T2[x] + T3[x]; }
}

// ---------------------------------------------------------------------------
extern "C" void kernel_launch(void* const* d_in, const int* in_sizes, int n_in,
                              void* d_out, int out_size, void* d_ws, size_t ws_size,
                              hipStream_t stream)
{
    (void)in_sizes; (void)n_in; (void)out_size; (void)ws_size;

    const float* path = (const float*)d_in[0];
    float* ws = (float*)d_ws;

    float* chunkSig = ws;                         // 256 * 4680 f32  (~4.8 MB)
    float* Pall     = chunkSig + NCHUNK * SIGLEN; // 256 * 584 f32
    float* T1 = Pall + NCHUNK * PLEN;             // 4096 (L4: P3 x A1)
    float* T2 = T1 + 4096;                        // 4096 (L4: P2 x A2)
    float* T3 = T2 + 4096;                        // 4096 (L4: P1 x A3)
    float* T4 = T3 + 4096;                        //  512 (L3: P2 x A1)
    float* T5 = T4 + 512;                         //  512 (L3: P1 x A2)
    float* T6 = T5 + 512;                         //   64 (L2: P1 x A1)
    float* out = (float*)d_out;

    sig_chunk_kernel<<<NCHUNK, 512, 0, stream>>>(path, chunkSig);
    prefix_scan_kernel<<<1, 256, 0, stream>>>(chunkSig, Pall);
    combine_wmma_kernel<<<89, 32, 0, stream>>>(Pall, chunkSig, T1, T2, T3, T4, T5, T6);
    finalize_kernel<<<(SIGLEN + 255) / 256, 256, 0, stream>>>(
        chunkSig, T1, T2, T3, T4, T5, T6, out);
}